// GraphCNN_41549513621585
// MI455X (gfx1250) — compile-verified
//
#include <hip/hip_runtime.h>
#include <hip/hip_bf16.h>
#include <math.h>

#define FEAT 128
#define HID  64

typedef __attribute__((ext_vector_type(2))) float v2f;
typedef __attribute__((ext_vector_type(4))) float v4f;
typedef __attribute__((ext_vector_type(8))) float v8f;

// ---------------- degree / norm ----------------
__global__ void deg_init(float* deg, int n) {
    int i = blockIdx.x * blockDim.x + threadIdx.x;
    if (i < n) deg[i] = 1.0f;              // self-loop contributes 1
}

__global__ void deg_count(const int* __restrict__ dst, float* deg, int E) {
    int e = blockIdx.x * blockDim.x + threadIdx.x;
    if (e < E) atomicAdd(&deg[dst[e]], 1.0f);
}

__global__ void deg_rsqrt(float* deg, int n) {
    int i = blockIdx.x * blockDim.x + threadIdx.x;
    if (i < n) deg[i] = rsqrtf(deg[i]);    // deg >= 1 always (self-loop)
}

// ---------------- fp32 WMMA GEMM: C[M,64] = A[M,K] @ B[K,64] ----------------
// One wave per 16-row tile; 8 waves (256 thr) per block -> 128 rows per block.
// B (<=32KB) staged in LDS once per block; inner loop uses ds_load for B,
// keeping the VMEM pipe free for streaming A rows.
__global__ __launch_bounds__(256)
void gemm_wmma(const float* __restrict__ A, const float* __restrict__ B,
               float* __restrict__ C, int M, int K) {
    __shared__ float Bs[FEAT * HID];       // worst case K=128 -> 32KB
    for (int i = threadIdx.x; i < K * HID; i += 256) Bs[i] = B[i];
    __syncthreads();

    int wave = threadIdx.x >> 5;
    int lane = threadIdx.x & 31;
    int m0 = (blockIdx.x * 8 + wave) * 16;
    if (m0 >= M) return;                   // wave-uniform exit, EXEC stays full
    int lh  = lane >> 4;                   // lane half (0/1)
    int l16 = lane & 15;

    int arow = m0 + l16;
    if (arow >= M) arow = M - 1;           // clamp tail rows (stores are guarded)
    const float* Ar = A + (long long)arow * K;

    v8f acc[4] = {};                       // 4 N-tiles of 16 -> full 64 cols
    for (int k0 = 0; k0 < K; k0 += 4) {
        int kk = k0 + 2 * lh;
        // A fragment: 16x4 f32, lane half selects K pair
        v2f a = *(const v2f*)(Ar + kk);
        const float* Bp = Bs + kk * HID + l16;
#pragma unroll
        for (int t = 0; t < 4; ++t) {
            v2f b;
            b.x = Bp[t * 16];              // (K=kk  , N=t*16+l16)
            b.y = Bp[t * 16 + HID];        // (K=kk+1, N=t*16+l16)
            acc[t] = __builtin_amdgcn_wmma_f32_16x16x4_f32(
                false, a, false, b, (short)0, acc[t], false, false);
        }
    }
#pragma unroll
    for (int t = 0; t < 4; ++t) {
#pragma unroll
        for (int r = 0; r < 8; ++r) {
            int row = m0 + 8 * lh + r;     // C layout: vgpr r -> M = r + 8*lh
            if (row < M)
                C[(long long)row * HID + t * 16 + l16] = acc[t][r];
        }
    }
}

// ---------------- aggregation ----------------
// out[i, 4f..4f+3] = h*dinv[i]^2 + bias   (self-loop term + bias, pre-scatter)
__global__ void selfloop_bias(const float* __restrict__ h, const float* __restrict__ dinv,
                              const float* __restrict__ bias, float* __restrict__ out, int n) {
    long long idx = (long long)blockIdx.x * blockDim.x + threadIdx.x;   // n*16
    if (idx >= (long long)n * 16) return;
    int i  = (int)(idx >> 4);
    int c  = (int)(idx & 15) * 4;
    float di = dinv[i];
    float d2 = di * di;
    v4f hv = *(const v4f*)(h + (long long)i * HID + c);
    v4f bv = *(const v4f*)(bias + c);
    v4f o;
    o.x = hv.x * d2 + bv.x;
    o.y = hv.y * d2 + bv.y;
    o.z = hv.z * d2 + bv.z;
    o.w = hv.w * d2 + bv.w;
    *(v4f*)(out + (long long)i * HID + c) = o;
}

// out[d,f] += h[s,f]*dinv[s]*dinv[d] -- float4 gather, coalesced f32 atomics
__global__ void edge_scatter(const float* __restrict__ h, const int* __restrict__ src,
                             const int* __restrict__ dst, const float* __restrict__ dinv,
                             float* __restrict__ out, int E) {
    long long idx = (long long)blockIdx.x * blockDim.x + threadIdx.x;   // E*16
    if (idx >= (long long)E * 16) return;
    int e = (int)(idx >> 4);
    int c = (int)(idx & 15) * 4;
    int s = src[e];
    int d = dst[e];
    float norm = dinv[s] * dinv[d];
    v4f hv = *(const v4f*)(h + (long long)s * HID + c);
    float* op = out + (long long)d * HID + c;
    atomicAdd(op + 0, hv.x * norm);
    atomicAdd(op + 1, hv.y * norm);
    atomicAdd(op + 2, hv.z * norm);
    atomicAdd(op + 3, hv.w * norm);
}

__global__ void relu_kernel(float* p, long long n4) {                   // n4 = n/4
    long long idx = (long long)blockIdx.x * blockDim.x + threadIdx.x;
    if (idx >= n4) return;
    v4f v = *(v4f*)(p + idx * 4);
    v.x = fmaxf(v.x, 0.0f);
    v.y = fmaxf(v.y, 0.0f);
    v.z = fmaxf(v.z, 0.0f);
    v.w = fmaxf(v.w, 0.0f);
    *(v4f*)(p + idx * 4) = v;
}

// ---------------- pooling ----------------
__global__ void pool_init(float* gsum, float* gmax, float* gcnt, int G) {
    int idx = blockIdx.x * blockDim.x + threadIdx.x;
    if (idx < G * HID) {
        gsum[idx] = 0.0f;
        ((int*)gmax)[idx] = (int)0xFF800000u;   // -inf
        if ((idx & 63) == 0) gcnt[idx >> 6] = 0.0f;
    }
}

__global__ void pool_scatter(const float* __restrict__ h, const int* __restrict__ batch,
                             float* gsum, float* gmax, float* gcnt, int n) {
    long long idx = (long long)blockIdx.x * blockDim.x + threadIdx.x;   // n*16
    if (idx >= (long long)n * 16) return;
    int i = (int)(idx >> 4);
    int c = (int)(idx & 15) * 4;
    int g = batch[i];
    v4f v = *(const v4f*)(h + (long long)i * HID + c);   // post-ReLU => v >= 0
    float* sp = gsum + g * HID + c;
    int*   mp = (int*)(gmax + g * HID + c);
    atomicAdd(sp + 0, v.x); atomicMax(mp + 0, __float_as_int(v.x));
    atomicAdd(sp + 1, v.y); atomicMax(mp + 1, __float_as_int(v.y));
    atomicAdd(sp + 2, v.z); atomicMax(mp + 2, __float_as_int(v.z));
    atomicAdd(sp + 3, v.w); atomicMax(mp + 3, __float_as_int(v.w));
    if (c == 0) atomicAdd(&gcnt[g], 1.0f);
}

// ---------------- MLP head: one 64-thread block per graph ----------------
__global__ __launch_bounds__(64)
void head_kernel(const float* __restrict__ gsum, const float* __restrict__ gmax,
                 const float* __restrict__ gcnt,
                 const float* __restrict__ fcW1, const float* __restrict__ fcb1,
                 const float* __restrict__ fcW2, const float* __restrict__ fcb2,
                 float* __restrict__ out) {
    __shared__ float g[2 * HID];
    __shared__ float z[HID];
    int gi = blockIdx.x;
    int t  = threadIdx.x;
    float cnt = gcnt[gi];
    cnt = cnt > 1.0f ? cnt : 1.0f;
    g[t]       = gsum[gi * HID + t] / cnt;       // mean
    g[HID + t] = gmax[gi * HID + t];             // max
    __syncthreads();
    float acc = fcb1[t];
    for (int k = 0; k < 2 * HID; ++k) acc += g[k] * fcW1[k * HID + t];
    z[t] = acc > 0.0f ? acc : 0.0f;
    __syncthreads();
    if (t == 0) {
        float s = fcb2[0];
        for (int j = 0; j < HID; ++j) s += z[j] * fcW2[j];
        out[gi] = 1.0f / (1.0f + expf(-s));
    }
}

extern "C" void kernel_launch(void* const* d_in, const int* in_sizes, int n_in,
                              void* d_out, int out_size, void* d_ws, size_t ws_size,
                              hipStream_t stream) {
    const float* x    = (const float*)d_in[0];
    const int*   ei   = (const int*)d_in[1];    // [2,E], src then dst
    const int*   batch= (const int*)d_in[2];
    const float* W1   = (const float*)d_in[3];
    const float* b1   = (const float*)d_in[4];
    const float* W2   = (const float*)d_in[5];
    const float* b2   = (const float*)d_in[6];
    const float* fcW1 = (const float*)d_in[7];
    const float* fcb1 = (const float*)d_in[8];
    const float* fcW2 = (const float*)d_in[9];
    const float* fcb2 = (const float*)d_in[10];
    float* out = (float*)d_out;

    int N = in_sizes[0] / FEAT;
    int E = in_sizes[1] / 2;
    int G = out_size;
    const int* src = ei;
    const int* dst = ei + E;

    float* ws = (float*)d_ws;
    size_t off = 0;
    float* dinv = ws + off; off += (size_t)((N + 255) & ~255);
    float* bufA = ws + off; off += (size_t)N * HID;
    float* bufB = ws + off; off += (size_t)N * HID;
    float* gsum = ws + off; off += (size_t)G * HID;
    float* gmax = ws + off; off += (size_t)G * HID;
    float* gcnt = ws + off; off += (size_t)G;

    dim3 blk(256);
    int nb_nodes = (N + 255) / 256;
    long long nVec = (long long)N * 16;          // float4 elements per node pass
    long long eVec = (long long)E * 16;
    long long nElem4 = (long long)N * HID / 4;
    int nb_vec  = (int)((nVec + 255) / 256);
    int nb_edge = (int)((eVec + 255) / 256);
    int nb_rel  = (int)((nElem4 + 255) / 256);
    int nb_gemm = (N + 127) / 128;

    // symmetric degree norm (shared by both layers)
    deg_init <<<nb_nodes, blk, 0, stream>>>(dinv, N);
    deg_count<<<(E + 255) / 256, blk, 0, stream>>>(dst, dinv, E);
    deg_rsqrt<<<nb_nodes, blk, 0, stream>>>(dinv, N);

    // layer 1: h = x@W1 ; agg = D^-1/2 A D^-1/2 h + b1 ; relu
    gemm_wmma    <<<nb_gemm, blk, 0, stream>>>(x, W1, bufA, N, FEAT);
    selfloop_bias<<<nb_vec,  blk, 0, stream>>>(bufA, dinv, b1, bufB, N);
    edge_scatter <<<nb_edge, blk, 0, stream>>>(bufA, src, dst, dinv, bufB, E);
    relu_kernel  <<<nb_rel,  blk, 0, stream>>>(bufB, nElem4);

    // layer 2
    gemm_wmma    <<<nb_gemm, blk, 0, stream>>>(bufB, W2, bufA, N, HID);
    selfloop_bias<<<nb_vec,  blk, 0, stream>>>(bufA, dinv, b2, bufB, N);
    edge_scatter <<<nb_edge, blk, 0, stream>>>(bufA, src, dst, dinv, bufB, E);
    relu_kernel  <<<nb_rel,  blk, 0, stream>>>(bufB, nElem4);

    // pooling + head
    pool_init   <<<(G * HID + 255) / 256, blk, 0, stream>>>(gsum, gmax, gcnt, G);
    pool_scatter<<<nb_vec,  blk, 0, stream>>>(bufB, batch, gsum, gmax, gcnt, N);
    head_kernel <<<G, dim3(64), 0, stream>>>(gsum, gmax, gcnt, fcW1, fcb1, fcW2, fcb2, out);
}